// RelativeMultiheadAttention_14851996909993
// MI455X (gfx1250) — compile-verified
//
#include <hip/hip_runtime.h>
#include <hip/hip_bf16.h>

typedef __attribute__((ext_vector_type(2))) float v2f;
typedef __attribute__((ext_vector_type(4))) float v4f;
typedef __attribute__((ext_vector_type(8))) float v8f;

constexpr int BB = 2, LL = 2048, EE = 1024, HH = 16, HD = 64;
constexpr int MM = BB * LL;           // 4096 rows in the big GEMMs
constexpr int NC = 16, CH = 128;      // cumsum chunking: 16 chunks of 128
constexpr float INVSCALE = 1.0f / 32.0f;  // 1/sqrt(E)

// ---------------- column sum over sequence dim: out[b,e] = sum_l in[b,l,e] ----
__global__ void colsum_kernel(const float* __restrict__ in, float* __restrict__ out) {
  int idx = blockIdx.x * blockDim.x + threadIdx.x;   // b*EE + e
  int b = idx / EE, e = idx % EE;
  const float* p = in + (size_t)b * LL * EE + e;
  float s0 = 0.f, s1 = 0.f, s2 = 0.f, s3 = 0.f;
  for (int l = 0; l < LL; l += 4) {
    s0 += p[(size_t)(l + 0) * EE];
    s1 += p[(size_t)(l + 1) * EE];
    s2 += p[(size_t)(l + 2) * EE];
    s3 += p[(size_t)(l + 3) * EE];
  }
  out[idx] = (s0 + s1) + (s2 + s3);
}

// -------- tiny vec-mat: out[b,n] = dot(x[b,:], W[n,:]) + biasMul*bias[n] -----
__global__ void vecmat_kernel(const float* __restrict__ x, const float* __restrict__ W,
                              const float* __restrict__ bias, float biasMul,
                              float* __restrict__ out) {
  int idx = blockIdx.x * blockDim.x + threadIdx.x;   // b*EE + n
  int b = idx / EE, n = idx % EE;
  const float* xr = x + (size_t)b * EE;
  const float* wr = W + (size_t)n * EE;
  float s = 0.f;
  for (int k = 0; k < EE; k += 4) {
    v4f xv = *(const v4f*)(xr + k);
    v4f wv = *(const v4f*)(wr + k);
    s += xv.x * wv.x + xv.y * wv.y + xv.z * wv.z + xv.w * wv.w;
  }
  out[idx] = s + bias[n] * biasMul;
}

// ---------------- WMMA FP32 GEMM: C[M,N] = A[M,K] @ W[N,K]^T (+ bias) --------
constexpr int GBM = 64, GBN = 64, GBK = 32, LDSS = 36;  // padded LDS row stride

__global__ __launch_bounds__(128)
void wmma_gemm_nt_kernel(const float* __restrict__ A, const float* __restrict__ W,
                         const float* __restrict__ bias, float* __restrict__ C,
                         int M, int N, int K) {
  __shared__ float As[GBM * LDSS];
  __shared__ float Bs[GBN * LDSS];
  const int tid  = threadIdx.x;
  const int wave = tid >> 5;
  const int lane = tid & 31;
  const int l16  = lane & 15;
  const int lhi  = lane >> 4;          // 0 or 1 -> K +0 / +2 half
  const int m0 = blockIdx.y * GBM;
  const int n0 = blockIdx.x * GBN;

  v8f acc0 = {}, acc1 = {}, acc2 = {}, acc3 = {};

  for (int k0 = 0; k0 < K; k0 += GBK) {
    __syncthreads();
    // cooperative stage: 64 rows x 32 cols each of A-tile and W-tile
#pragma unroll
    for (int i = 0; i < 4; ++i) {
      int idx = tid + (i << 7);        // 0..511 float4 slots
      int row = idx >> 3;
      int c4  = (idx & 7) << 2;
      *(v4f*)(As + row * LDSS + c4) = *(const v4f*)(A + (size_t)(m0 + row) * K + k0 + c4);
      *(v4f*)(Bs + row * LDSS + c4) = *(const v4f*)(W + (size_t)(n0 + row) * K + k0 + c4);
    }
    __syncthreads();

    const float* aBase = As + ((wave << 4) + l16) * LDSS;
    const float* bBase = Bs + l16 * LDSS;
#pragma unroll
    for (int kk = 0; kk < GBK; kk += 4) {
      int kf = kk + (lhi << 1);        // lanes 0-15: K=kk,kk+1 ; lanes 16-31: K=kk+2,kk+3
      v2f av = *(const v2f*)(aBase + kf);
      v2f b0 = *(const v2f*)(bBase + 0 * 16 * LDSS + kf);
      v2f b1 = *(const v2f*)(bBase + 1 * 16 * LDSS + kf);
      v2f b2 = *(const v2f*)(bBase + 2 * 16 * LDSS + kf);
      v2f b3 = *(const v2f*)(bBase + 3 * 16 * LDSS + kf);
      acc0 = __builtin_amdgcn_wmma_f32_16x16x4_f32(false, av, false, b0, (short)0, acc0, false, false);
      acc1 = __builtin_amdgcn_wmma_f32_16x16x4_f32(false, av, false, b1, (short)0, acc1, false, false);
      acc2 = __builtin_amdgcn_wmma_f32_16x16x4_f32(false, av, false, b2, (short)0, acc2, false, false);
      acc3 = __builtin_amdgcn_wmma_f32_16x16x4_f32(false, av, false, b3, (short)0, acc3, false, false);
    }
  }

  // epilogue: C/D layout — VGPR r: lanes 0-15 => M=r, lanes 16-31 => M=r+8
  const int mBase = m0 + (wave << 4) + (lhi << 3);
  float bb0 = bias ? bias[n0 +  0 + l16] : 0.f;
  float bb1 = bias ? bias[n0 + 16 + l16] : 0.f;
  float bb2 = bias ? bias[n0 + 32 + l16] : 0.f;
  float bb3 = bias ? bias[n0 + 48 + l16] : 0.f;
#pragma unroll
  for (int r = 0; r < 8; ++r) {
    float* crow = C + (size_t)(mBase + r) * N + n0;
    crow[ 0 + l16] = acc0[r] + bb0;
    crow[16 + l16] = acc1[r] + bb1;
    crow[32 + l16] = acc2[r] + bb2;
    crow[48 + l16] = acc3[r] + bb3;
  }
}

// ---------------- chunk sums of r over seq dim ------------------------------
__global__ void chunksum_kernel(const float* __restrict__ r, float* __restrict__ csum) {
  int idx = blockIdx.x * blockDim.x + threadIdx.x;   // b*NC*EE + j*EE + e
  int e = idx % EE; int t = idx / EE; int j = t % NC; int b = t / NC;
  const float* p = r + ((size_t)b * LL + (size_t)j * CH) * EE + e;
  float s0 = 0.f, s1 = 0.f, s2 = 0.f, s3 = 0.f;
  for (int c = 0; c < CH; c += 4) {
    s0 += p[(size_t)(c + 0) * EE];
    s1 += p[(size_t)(c + 1) * EE];
    s2 += p[(size_t)(c + 2) * EE];
    s3 += p[(size_t)(c + 3) * EE];
  }
  csum[idx] = (s0 + s1) + (s2 + s3);
}

// ---- in-place: csum[b,j,e] -> exclusive suffix sum over chunks (offset) ----
__global__ void chunkscan_kernel(float* __restrict__ csum) {
  int idx = blockIdx.x * blockDim.x + threadIdx.x;   // b*EE + e
  int b = idx / EE, e = idx % EE;
  float vals[NC];
#pragma unroll
  for (int j = 0; j < NC; ++j) vals[j] = csum[((size_t)b * NC + j) * EE + e];
  float acc = 0.f;
#pragma unroll
  for (int j = NC - 1; j >= 0; --j) {
    csum[((size_t)b * NC + j) * EE + e] = acc;  // exclusive: sum of chunks > j
    acc += vals[j];
  }
}

// ---- in-place suffix cumsum within each chunk: r becomes Rsuf --------------
__global__ void suffix_inplace_kernel(float* __restrict__ r, const float* __restrict__ coff) {
  int idx = blockIdx.x * blockDim.x + threadIdx.x;   // b*NC*EE + j*EE + e
  int e = idx % EE; int t = idx / EE; int j = t % NC; int b = t / NC;
  float acc = coff[idx];
  float* base = r + ((size_t)b * LL + (size_t)j * CH) * EE + e;
  for (int c = CH - 4; c >= 0; c -= 4) {
    float v3 = base[(size_t)(c + 3) * EE];
    float v2 = base[(size_t)(c + 2) * EE];
    float v1 = base[(size_t)(c + 1) * EE];
    float v0 = base[(size_t)(c + 0) * EE];
    acc += v3; base[(size_t)(c + 3) * EE] = acc;
    acc += v2; base[(size_t)(c + 2) * EE] = acc;
    acc += v1; base[(size_t)(c + 1) * EE] = acc;
    acc += v0; base[(size_t)(c + 0) * EE] = acc;
  }
}

// ---- score row-sums S[b,h,q] -----------------------------------------------
__global__ void score_kernel(const float* __restrict__ q, const float* __restrict__ Ksum,
                             const float* __restrict__ Rsuf, const float* __restrict__ U,
                             const float* __restrict__ V, float* __restrict__ S) {
  int idx = blockIdx.x * blockDim.x + threadIdx.x;   // b*HH*LL + h*LL + qi
  int qi = idx % LL; int t = idx / LL; int h = t % HH; int b = t / HH;
  const float* qrow = q + ((size_t)b * LL + qi) * EE + h * HD;
  const float* ks   = Ksum + ((size_t)b * EE) + h * HD;
  const float* u    = U + h * HD;
  const float* v    = V + h * HD;
  const float* rsf  = Rsuf + ((size_t)b * LL + (LL - 1 - qi)) * EE + h * HD;
  float s = 0.f;
#pragma unroll
  for (int d = 0; d < HD; d += 4) {
    v4f qv = *(const v4f*)(qrow + d);
    v4f uu = *(const v4f*)(u + d);
    v4f vv = *(const v4f*)(v + d);
    v4f kk = *(const v4f*)(ks + d);
    v4f rr = *(const v4f*)(rsf + d);
    s += (qv.x + uu.x) * kk.x + (qv.y + uu.y) * kk.y + (qv.z + uu.z) * kk.z + (qv.w + uu.w) * kk.w;
    s += (qv.x + vv.x) * rr.x + (qv.y + vv.y) * rr.y + (qv.z + vv.z) * rr.z + (qv.w + vv.w) * rr.w;
  }
  if (qi <= LL - 3) {  // prefix term: row q+1, cols [0, LL-qi-3] = Rsuf[0]-Rsuf[LL-qi-2]
    const float* q2  = q + ((size_t)b * LL + qi + 1) * EE + h * HD;
    const float* r0  = Rsuf + ((size_t)b * LL + 0) * EE + h * HD;
    const float* rm  = Rsuf + ((size_t)b * LL + (LL - qi - 2)) * EE + h * HD;
#pragma unroll
    for (int d = 0; d < HD; d += 4) {
      v4f qv = *(const v4f*)(q2 + d);
      v4f vv = *(const v4f*)(v + d);
      v4f a  = *(const v4f*)(r0 + d);
      v4f bvv = *(const v4f*)(rm + d);
      s += (qv.x + vv.x) * (a.x - bvv.x) + (qv.y + vv.y) * (a.y - bvv.y)
         + (qv.z + vv.z) * (a.z - bvv.z) + (qv.w + vv.w) * (a.w - bvv.w);
    }
  }
  S[idx] = s * INVSCALE;
}

// ---- W2[b,o,h] = dot(Vsum[b, h*HD: ], Wo[o, h*HD: ]) -----------------------
__global__ void w2_kernel(const float* __restrict__ Vsum, const float* __restrict__ Wo,
                          float* __restrict__ W2) {
  int idx = blockIdx.x * blockDim.x + threadIdx.x;   // b*EE*HH + o*HH + h
  int h = idx % HH; int t = idx / HH; int o = t % EE; int b = t / EE;
  const float* vs = Vsum + (size_t)b * EE + h * HD;
  const float* wr = Wo + (size_t)o * EE + h * HD;
  float s = 0.f;
#pragma unroll
  for (int d = 0; d < HD; d += 4) {
    v4f a = *(const v4f*)(vs + d);
    v4f w = *(const v4f*)(wr + d);
    s += a.x * w.x + a.y * w.y + a.z * w.z + a.w * w.w;
  }
  W2[idx] = s;
}

// ---- out[b,q,o] = bo[o] + sum_h S[b,h,q] * W2[b,o,h] -----------------------
__global__ void final_kernel(const float* __restrict__ S, const float* __restrict__ W2,
                             const float* __restrict__ bo, float* __restrict__ out) {
  int idx = blockIdx.x * blockDim.x + threadIdx.x;   // b*LL*EE + qi*EE + o
  int o = idx % EE; int t = idx / EE; int qi = t % LL; int b = t / LL;
  const float* w2r = W2 + ((size_t)b * EE + o) * HH;
  float acc = bo[o];
#pragma unroll
  for (int h = 0; h < HH; ++h)
    acc += S[((size_t)b * HH + h) * LL + qi] * w2r[h];
  out[idx] = acc;
}

extern "C" void kernel_launch(void* const* d_in, const int* in_sizes, int n_in,
                              void* d_out, int out_size, void* d_ws, size_t ws_size,
                              hipStream_t stream) {
  const float* query = (const float*)d_in[0];
  const float* key   = (const float*)d_in[1];
  const float* value = (const float*)d_in[2];
  const float* pos   = (const float*)d_in[3];
  const float* Wq = (const float*)d_in[4];
  const float* bq = (const float*)d_in[5];
  const float* Wk = (const float*)d_in[6];
  const float* bk = (const float*)d_in[7];
  const float* Wv = (const float*)d_in[8];
  const float* bv = (const float*)d_in[9];
  const float* Wp = (const float*)d_in[10];
  const float* Wo = (const float*)d_in[11];
  const float* bo = (const float*)d_in[12];
  const float* U  = (const float*)d_in[13];
  const float* V  = (const float*)d_in[14];
  float* out = (float*)d_out;

  float* ws   = (float*)d_ws;
  float* qb   = ws;                              // [MM,EE] projected q
  float* rb   = qb + (size_t)MM * EE;            // [MM,EE] r -> suffix-cumsum in place
  float* ksum = rb + (size_t)MM * EE;            // [BB,EE]
  float* vsum = ksum + BB * EE;                  // [BB,EE]
  float* krow = vsum + BB * EE;                  // [BB,EE] raw key colsum
  float* vrow = krow + BB * EE;                  // [BB,EE] raw value colsum
  float* csum = vrow + BB * EE;                  // [BB,NC,EE]
  float* Sbuf = csum + (size_t)BB * NC * EE;     // [BB,HH,LL]
  float* W2   = Sbuf + (size_t)BB * HH * LL;     // [BB,EE,HH]

  colsum_kernel<<<(BB * EE) / 256, 256, 0, stream>>>(key, krow);
  colsum_kernel<<<(BB * EE) / 256, 256, 0, stream>>>(value, vrow);
  vecmat_kernel<<<(BB * EE) / 256, 256, 0, stream>>>(krow, Wk, bk, (float)LL, ksum);
  vecmat_kernel<<<(BB * EE) / 256, 256, 0, stream>>>(vrow, Wv, bv, (float)LL, vsum);

  dim3 ggrid(EE / GBN, MM / GBM);
  wmma_gemm_nt_kernel<<<ggrid, 128, 0, stream>>>(query, Wq, bq, qb, MM, EE, EE);
  wmma_gemm_nt_kernel<<<ggrid, 128, 0, stream>>>(pos, Wp, nullptr, rb, MM, EE, EE);

  chunksum_kernel<<<(BB * NC * EE) / 256, 256, 0, stream>>>(rb, csum);
  chunkscan_kernel<<<(BB * EE) / 256, 256, 0, stream>>>(csum);
  suffix_inplace_kernel<<<(BB * NC * EE) / 256, 256, 0, stream>>>(rb, csum);

  score_kernel<<<(BB * HH * LL) / 256, 256, 0, stream>>>(qb, ksum, rb, U, V, Sbuf);
  w2_kernel<<<(BB * EE * HH) / 256, 256, 0, stream>>>(vsum, Wo, W2);
  final_kernel<<<(BB * LL * EE) / 256, 256, 0, stream>>>(Sbuf, W2, bo, out);
}